// DiffFDN_76510547411609
// MI455X (gfx1250) — compile-verified
//
#include <hip/hip_runtime.h>
#include <math.h>

// ---------------------------------------------------------------------------
// DiffFDN on MI455X (gfx1250):
//   K1 setup      : expm(skew(W)) -> A, AG = A*diag(gamma), md  (1 thread)
//   K2 solve      : per-frequency 8x8 complex GE solve, H + Hsum (1 lane/freq)
//   K3 spectrum   : Hermitian-extend Hsum -> X matrix [960 x 400]
//   K4 twiddles   : DFT matrices E2(400x400), E1(960x960), twiddle T(960x400)
//   K5 cgemm+tw   : Y' = (X @ E2) .* T         -- v_wmma_f32_16x16x4_f32
//   K6 rgemm      : h  = Re(E1 @ Y') / NFFT    -- v_wmma_f32_16x16x4_f32
//   K7 maxabs, K8 normalize
// Four-step factorization: NFFT = 384000 = 960*400, k = k1 + 960*k2,
// n = n2 + 400*n1  =>  h row-major [n1][n2] is already flat h[n].
// ---------------------------------------------------------------------------

typedef __attribute__((ext_vector_type(2))) float v2f;
typedef __attribute__((ext_vector_type(8))) float v8f;

#define NCH   8
#define NF    192001
#define NFFTL 384000
#define DN1   960
#define DN2   400

// workspace layout (float offsets); total ~5.24M floats (~21 MB)
static const size_t O_AG   = 0;        // 64
static const size_t O_MD   = 64;       // 8
static const size_t O_MAX  = 72;       // 1 (uint bits)
static const size_t O_HSUM = 128;      // 2*NF
static const size_t O_XR   = 384256;   // 384000
static const size_t O_XI   = 768256;   // 384000
static const size_t O_E2R  = 1152256;  // 160000
static const size_t O_E2I  = 1312256;  // 160000
static const size_t O_E1R  = 1472256;  // 921600
static const size_t O_E1I  = 2393856;  // 921600
static const size_t O_TR   = 3315456;  // 384000
static const size_t O_TI   = 3699456;  // 384000
static const size_t O_YR   = 4083456;  // 384000
static const size_t O_YI   = 4467456;  // 384000
static const size_t O_HM   = 4851456;  // 384000

// ----------------------------- complex helpers -----------------------------
struct cpx { float re, im; };
__device__ __forceinline__ cpx cmul(cpx a, cpx b) {
  cpx r; r.re = a.re * b.re - a.im * b.im; r.im = a.re * b.im + a.im * b.re; return r;
}
__device__ __forceinline__ cpx csub(cpx a, cpx b) { cpx r{a.re - b.re, a.im - b.im}; return r; }
__device__ __forceinline__ cpx crecip(cpx a) {
  float d = a.re * a.re + a.im * a.im; float id = 1.0f / d;
  cpx r{a.re * id, -a.im * id}; return r;
}

// ----------------------------- WMMA wrapper --------------------------------
__device__ __forceinline__ v8f wmma_f32(v2f a, v2f b, v8f c) {
  // D = A(16x4 f32) x B(4x16 f32) + C(16x16 f32)
  return __builtin_amdgcn_wmma_f32_16x16x4_f32(false, a, false, b, (short)0, c,
                                               false, false);
}
// A fragment: row0..row0+15 x K{k0..k0+3}; lanes 0-15 hold K{k0,k0+1}, 16-31 K{k0+2,k0+3}
__device__ __forceinline__ v2f fragA(const float* __restrict__ A, int lda,
                                     int row0, int k0, int lr, int half) {
  int kk = k0 + 2 * half;
  const float* p = A + (size_t)(row0 + lr) * lda + kk;
  v2f a; a.x = p[0]; a.y = p[1]; return a;
}
// B fragment: K{k0..k0+3} x col0..col0+15; lanes 0-15 rows K{k0,k0+1}, 16-31 K{k0+2,k0+3}
__device__ __forceinline__ v2f fragB(const float* __restrict__ B, int ldb,
                                     int col0, int k0, int lr, int half) {
  int kk = k0 + 2 * half;
  const float* p = B + (size_t)kk * ldb + col0 + lr;
  v2f b; b.x = p[0]; b.y = p[ldb]; return b;
}

// ------------------------ K1: expm + gains (1 thread) ----------------------
__global__ void fdn_setup_kernel(const float* __restrict__ W,
                                 const float* __restrict__ m,
                                 float* __restrict__ ws) {
  if (threadIdx.x != 0 || blockIdx.x != 0) return;
  double K[NCH][NCH], E[NCH][NCH], T[NCH][NCH];
  // K = triu(W,1) - triu(W,1)^T
  for (int i = 0; i < NCH; ++i)
    for (int j = 0; j < NCH; ++j) {
      double s  = (j > i) ? (double)W[i * NCH + j] : 0.0;
      double st = (i > j) ? (double)W[j * NCH + i] : 0.0;
      K[i][j] = s - st;
    }
  // scale so ||K||_inf / 2^s <= 0.25
  double nrm = 0.0;
  for (int i = 0; i < NCH; ++i) {
    double r = 0.0;
    for (int j = 0; j < NCH; ++j) r += fabs(K[i][j]);
    nrm = fmax(nrm, r);
  }
  int s = 0;
  while (nrm > 0.25 && s < 40) { nrm *= 0.5; s++; }
  double sc = 1.0; for (int q = 0; q < s; ++q) sc *= 0.5;
  for (int i = 0; i < NCH; ++i)
    for (int j = 0; j < NCH; ++j) K[i][j] *= sc;
  // Taylor (Horner, order 16): E = I + K(I + K/2(I + ...))
  for (int i = 0; i < NCH; ++i)
    for (int j = 0; j < NCH; ++j) E[i][j] = (i == j) ? 1.0 : 0.0;
  for (int t = 16; t >= 1; --t) {
    for (int i = 0; i < NCH; ++i)
      for (int j = 0; j < NCH; ++j) {
        double acc = 0.0;
        for (int q = 0; q < NCH; ++q) acc += K[i][q] * E[q][j];
        T[i][j] = (i == j ? 1.0 : 0.0) + acc / (double)t;
      }
    for (int i = 0; i < NCH; ++i)
      for (int j = 0; j < NCH; ++j) E[i][j] = T[i][j];
  }
  // square s times
  for (int q = 0; q < s; ++q) {
    for (int i = 0; i < NCH; ++i)
      for (int j = 0; j < NCH; ++j) {
        double acc = 0.0;
        for (int p = 0; p < NCH; ++p) acc += E[i][p] * E[p][j];
        T[i][j] = acc;
      }
    for (int i = 0; i < NCH; ++i)
      for (int j = 0; j < NCH; ++j) E[i][j] = T[i][j];
  }
  // de-normalize delays; gamma = 0.9999^md ; AG = A * diag(gamma)
  const double delays[NCH] = {809., 877., 937., 1049., 1151., 1249., 1373., 1499.};
  double avg = 0.0;
  for (int i = 0; i < NCH; ++i) avg += delays[i];
  avg /= (double)NCH;
  double var = 0.0;
  for (int i = 0; i < NCH; ++i) { double d = delays[i] - avg; var += d * d; }
  double stdv = sqrt(var / (double)(NCH - 1));
  double lg = log(0.9999);
  double gam[NCH];
  for (int n = 0; n < NCH; ++n) {
    double md = (double)m[n] * stdv + avg;
    ws[O_MD + n] = (float)md;
    gam[n] = exp(md * lg);
  }
  for (int i = 0; i < NCH; ++i)
    for (int j = 0; j < NCH; ++j)
      ws[O_AG + i * NCH + j] = (float)(E[i][j] * gam[j]);
  ((unsigned int*)ws)[O_MAX] = 0u;  // reset for max-abs reduction
}

// --------------- K2: per-frequency 8x8 complex solve (1 lane/f) ------------
__global__ void fdn_solve_kernel(float* __restrict__ ws,
                                 const float* __restrict__ B,
                                 const float* __restrict__ C,
                                 float* __restrict__ Hout) {
  __shared__ float sAG[NCH * NCH], sMD[NCH], sB[NCH], sC[NCH];
  int t = threadIdx.x;
  if (t < NCH * NCH) sAG[t] = ws[O_AG + t];
  if (t < NCH) { sMD[t] = ws[O_MD + t]; sB[t] = B[t]; sC[t] = C[t]; }
  __syncthreads();
  int f = blockIdx.x * blockDim.x + t;
  if (f >= NF) return;

  const double w = 3.14159265358979323846 * (double)f / (double)(NF - 1);
  cpx M[NCH][NCH], rhs[NCH];
#pragma unroll
  for (int i = 0; i < NCH; ++i) {
#pragma unroll
    for (int j = 0; j < NCH; ++j) { M[i][j].re = -sAG[i * NCH + j]; M[i][j].im = 0.0f; }
    rhs[i].re = sB[i]; rhs[i].im = 0.0f;
  }
#pragma unroll
  for (int n = 0; n < NCH; ++n) {
    // z^md = e^{i * md * w}; reduce phase in double for large md*w
    double ang = fmod((double)sMD[n] * w, 6.28318530717958647692);
    double ds, dc; sincos(ang, &ds, &dc);
    M[n][n].re += (float)dc; M[n][n].im += (float)ds;
  }
  // Gaussian elimination (diag-normalized), fully unrolled -> registers
#pragma unroll
  for (int k = 0; k < NCH; ++k) {
    cpx inv = crecip(M[k][k]);
#pragma unroll
    for (int j = 0; j < NCH; ++j)
      if (j > k) M[k][j] = cmul(M[k][j], inv);
    rhs[k] = cmul(rhs[k], inv);
#pragma unroll
    for (int i = 0; i < NCH; ++i)
      if (i > k) {
        cpx fac = M[i][k];
#pragma unroll
        for (int j = 0; j < NCH; ++j)
          if (j > k) M[i][j] = csub(M[i][j], cmul(fac, M[k][j]));
        rhs[i] = csub(rhs[i], cmul(fac, rhs[k]));
      }
  }
#pragma unroll
  for (int k = NCH - 1; k >= 0; --k) {
    cpx acc = rhs[k];
#pragma unroll
    for (int j = 0; j < NCH; ++j)
      if (j > k) acc = csub(acc, cmul(M[k][j], rhs[j]));
    rhs[k] = acc;
  }
  // H[f][n] = x_n * C[n]; Hsum = sum_n
  float sr = 0.0f, si = 0.0f;
#pragma unroll
  for (int n = 0; n < NCH; ++n) {
    float hr = rhs[n].re * sC[n];
    float hi = rhs[n].im * sC[n];
    Hout[2 * ((size_t)f * NCH + n) + 0] = hr;
    Hout[2 * ((size_t)f * NCH + n) + 1] = hi;
    sr += hr; si += hi;
  }
  ws[O_HSUM + 2 * (size_t)f + 0] = sr;
  ws[O_HSUM + 2 * (size_t)f + 1] = si;
}

// ------- K3: Hermitian-extend Hsum into X matrix layout [k1=960][k2=400] ---
__global__ void fdn_spectrum_kernel(float* __restrict__ ws) {
  int k = blockIdx.x * blockDim.x + threadIdx.x;
  if (k >= NFFTL) return;
  float re, im;
  if (k <= NFFTL / 2) {
    re = ws[O_HSUM + 2 * (size_t)k + 0];
    im = ws[O_HSUM + 2 * (size_t)k + 1];
  } else {
    int km = NFFTL - k;
    re =  ws[O_HSUM + 2 * (size_t)km + 0];
    im = -ws[O_HSUM + 2 * (size_t)km + 1];
  }
  int k1 = k % DN1, k2 = k / DN1;
  ws[O_XR + (size_t)k1 * DN2 + k2] = re;
  ws[O_XI + (size_t)k1 * DN2 + k2] = im;
}

// -------- K4: DFT matrices E2 (400x400), E1 (960x960), twiddle T -----------
__global__ void fdn_twiddle_kernel(float* __restrict__ ws) {
  const double TWO_PI = 6.28318530717958647692;
  int stride = gridDim.x * blockDim.x;
  int tid = blockIdx.x * blockDim.x + threadIdx.x;
  for (int i = tid; i < DN1 * DN1; i += stride) {          // E1[n1][k1]
    int n1 = i / DN1, k1 = i % DN1;
    int ph = (n1 * k1) % DN1;
    float sn, cs; sincosf((float)(TWO_PI * (double)ph / (double)DN1), &sn, &cs);
    ws[O_E1R + i] = cs; ws[O_E1I + i] = sn;
  }
  for (int i = tid; i < DN2 * DN2; i += stride) {          // E2[k2][n2]
    int k2 = i / DN2, n2 = i % DN2;
    int ph = (k2 * n2) % DN2;
    float sn, cs; sincosf((float)(TWO_PI * (double)ph / (double)DN2), &sn, &cs);
    ws[O_E2R + i] = cs; ws[O_E2I + i] = sn;
  }
  for (int i = tid; i < DN1 * DN2; i += stride) {          // T[k1][n2]
    int k1 = i / DN2, n2 = i % DN2;
    long long ph = ((long long)k1 * n2) % NFFTL;
    float sn, cs; sincosf((float)(TWO_PI * (double)ph / (double)NFFTL), &sn, &cs);
    ws[O_TR + i] = cs; ws[O_TI + i] = sn;
  }
}

// ------ K5: complex GEMM Y = X @ E2, fused twiddle; one wave per 16x16 -----
__global__ __launch_bounds__(32) void fdn_cgemm_twiddle_kernel(float* __restrict__ ws) {
  const float* Xr  = ws + O_XR;  const float* Xi  = ws + O_XI;
  const float* E2r = ws + O_E2R; const float* E2i = ws + O_E2I;
  const float* Tr  = ws + O_TR;  const float* Ti  = ws + O_TI;
  float* Yr = ws + O_YR; float* Yi = ws + O_YI;
  int lane = threadIdx.x & 31, half = lane >> 4, lr = lane & 15;
  int col0 = blockIdx.x * 16;   // n2
  int row0 = blockIdx.y * 16;   // k1
  v8f arr = {}, aii = {}, ari = {}, air = {};
  for (int k0 = 0; k0 < DN2; k0 += 4) {
    v2f xr = fragA(Xr, DN2, row0, k0, lr, half);
    v2f xi = fragA(Xi, DN2, row0, k0, lr, half);
    v2f er = fragB(E2r, DN2, col0, k0, lr, half);
    v2f ei = fragB(E2i, DN2, col0, k0, lr, half);
    arr = wmma_f32(xr, er, arr);
    aii = wmma_f32(xi, ei, aii);
    ari = wmma_f32(xr, ei, ari);
    air = wmma_f32(xi, er, air);
  }
#pragma unroll
  for (int r = 0; r < 8; ++r) {
    int row = row0 + r + 8 * half, col = col0 + lr;
    size_t idx = (size_t)row * DN2 + col;
    float yr = arr[r] - aii[r];
    float yi = ari[r] + air[r];
    float tr = Tr[idx], ti = Ti[idx];
    Yr[idx] = yr * tr - yi * ti;
    Yi[idx] = yr * ti + yi * tr;
  }
}

// ----- K6: real-part GEMM h = Re(E1 @ Y')/NFFT; one wave per 16x16 tile ----
__global__ __launch_bounds__(32) void fdn_rgemm_kernel(float* __restrict__ ws) {
  const float* E1r = ws + O_E1R; const float* E1i = ws + O_E1I;
  const float* Yr  = ws + O_YR;  const float* Yi  = ws + O_YI;
  float* Hm = ws + O_HM;
  int lane = threadIdx.x & 31, half = lane >> 4, lr = lane & 15;
  int col0 = blockIdx.x * 16;   // n2
  int row0 = blockIdx.y * 16;   // n1
  v8f a1 = {}, a2 = {};
  for (int k0 = 0; k0 < DN1; k0 += 4) {
    v2f er = fragA(E1r, DN1, row0, k0, lr, half);
    v2f ei = fragA(E1i, DN1, row0, k0, lr, half);
    v2f yr = fragB(Yr, DN2, col0, k0, lr, half);
    v2f yi = fragB(Yi, DN2, col0, k0, lr, half);
    a1 = wmma_f32(er, yr, a1);
    a2 = wmma_f32(ei, yi, a2);
  }
  const float inv_n = 1.0f / (float)NFFTL;
#pragma unroll
  for (int r = 0; r < 8; ++r) {
    int row = row0 + r + 8 * half, col = col0 + lr;
    Hm[(size_t)row * DN2 + col] = (a1[r] - a2[r]) * inv_n;  // == h[row*400+col]
  }
}

// --------------------------- K7: max |h| reduction -------------------------
__global__ void fdn_maxabs_kernel(float* __restrict__ ws) {
  __shared__ float sm[256];
  const float* Hm = ws + O_HM;
  unsigned int* pmax = (unsigned int*)ws + O_MAX;
  float v = 0.0f;
  for (int i = blockIdx.x * blockDim.x + threadIdx.x; i < NFFTL;
       i += gridDim.x * blockDim.x)
    v = fmaxf(v, fabsf(Hm[i]));
  sm[threadIdx.x] = v;
  __syncthreads();
  for (int s = 128; s > 0; s >>= 1) {
    if (threadIdx.x < s) sm[threadIdx.x] = fmaxf(sm[threadIdx.x], sm[threadIdx.x + s]);
    __syncthreads();
  }
  if (threadIdx.x == 0) atomicMax(pmax, __float_as_uint(sm[0]));
}

// --------------------------- K8: normalize h -------------------------------
__global__ void fdn_norm_kernel(const float* __restrict__ ws, float* __restrict__ hout) {
  float mx = __uint_as_float(((const unsigned int*)ws)[O_MAX]);
  float inv = 1.0f / (mx > 0.0f ? mx : 1.0f);
  const float* Hm = ws + O_HM;
  for (int i = blockIdx.x * blockDim.x + threadIdx.x; i < NFFTL;
       i += gridDim.x * blockDim.x)
    hout[i] = Hm[i] * inv;
}

// ---------------------------------------------------------------------------
extern "C" void kernel_launch(void* const* d_in, const int* in_sizes, int n_in,
                              void* d_out, int out_size, void* d_ws, size_t ws_size,
                              hipStream_t stream) {
  (void)in_sizes; (void)n_in; (void)out_size; (void)ws_size;
  // inputs: x (unused; x = e^{i*pi*f/(F-1)} recomputed), B[8], C[8], W[64], m[8]
  const float* B = (const float*)d_in[1];
  const float* C = (const float*)d_in[2];
  const float* W = (const float*)d_in[3];
  const float* m = (const float*)d_in[4];
  float* ws   = (float*)d_ws;
  float* Hout = (float*)d_out;                       // [F,8] complex64 interleaved
  float* hout = (float*)d_out + 2 * (size_t)NF * NCH; // [384000] float32

  fdn_setup_kernel<<<1, 64, 0, stream>>>(W, m, ws);
  fdn_solve_kernel<<<(NF + 255) / 256, 256, 0, stream>>>(ws, B, C, Hout);
  fdn_spectrum_kernel<<<(NFFTL + 255) / 256, 256, 0, stream>>>(ws);
  fdn_twiddle_kernel<<<1024, 256, 0, stream>>>(ws);
  dim3 g1(DN2 / 16, DN1 / 16);  // 25 x 60 tiles, one wave each
  fdn_cgemm_twiddle_kernel<<<g1, 32, 0, stream>>>(ws);
  fdn_rgemm_kernel<<<g1, 32, 0, stream>>>(ws);
  fdn_maxabs_kernel<<<256, 256, 0, stream>>>(ws);
  fdn_norm_kernel<<<(NFFTL + 255) / 256, 256, 0, stream>>>(ws, hout);
}